// Hin2vec_layer_26517128085717
// MI455X (gfx1250) — compile-verified
//
#include <hip/hip_runtime.h>
#include <hip/hip_bf16.h>
#include <math.h>

// Problem constants (match reference)
#define B_ROWS   262144
#define H_DIM    128
#define R_ROWS   64
#define WR_ELEMS (R_ROWS * H_DIM)      // 8192 floats = 32 KB

#define BLOCK    256                    // 8 waves (wave32)
#define WAVES_PER_BLOCK (BLOCK / 32)
#define NBLOCKS  1024                   // 8192 waves -> 32 rows/wave

typedef int v4i __attribute__((ext_vector_type(4)));
typedef __attribute__((address_space(1))) v4i* gptr_v4i;   // global
typedef __attribute__((address_space(3))) v4i* lptr_v4i;   // LDS

// ---------------------------------------------------------------------------
// Kernel 1: gather + fused sigmoid' + dot + logits, per-block loss partials.
// One wave32 per batch row; 32 lanes x float4 == H=128.
// ---------------------------------------------------------------------------
__global__ __launch_bounds__(BLOCK) void hin2vec_fwd(
    const int* __restrict__ x, const int* __restrict__ y,
    const int* __restrict__ r, const int* __restrict__ l,
    const float* __restrict__ Wx, const float* __restrict__ Wr,
    float* __restrict__ out, float* __restrict__ blockPartials)
{
    __shared__ float sWr[WR_ELEMS];                  // 32 KB staged Wr -> s*(1-s)
    __shared__ float sWaveLoss[WAVES_PER_BLOCK];

    const int tid = threadIdx.x;

    // ---- Stage Wr (32 KB) into LDS via async global->LDS DMA ----
#if __has_builtin(__builtin_amdgcn_global_load_async_to_lds_b128)
    {
        #pragma unroll
        for (int i = 0; i < (WR_ELEMS / 4) / BLOCK; ++i) {    // 2048 float4 / 256 thr = 8
            const int c = tid + i * BLOCK;                     // float4 chunk index
            __builtin_amdgcn_global_load_async_to_lds_b128(
                (gptr_v4i)(Wr + (size_t)c * 4),
                (lptr_v4i)(sWr + (size_t)c * 4),
                0, 0);
        }
#if __has_builtin(__builtin_amdgcn_s_wait_asynccnt)
        __builtin_amdgcn_s_wait_asynccnt(0);
#else
        asm volatile("s_wait_asynccnt 0" ::: "memory");
#endif
    }
#else
    for (int c = tid; c < WR_ELEMS / 4; c += BLOCK) {
        ((float4*)sWr)[c] = ((const float4*)Wr)[c];
    }
#endif
    __syncthreads();

    // ---- Precompute regularizer rows in LDS: re = s*(1-s), s=sigmoid(clamp) ----
    for (int i = tid; i < WR_ELEMS; i += BLOCK) {
        float v = sWr[i];
        v = fminf(fmaxf(v, -6.0f), 6.0f);
        const float s = 1.0f / (1.0f + expf(-v));
        sWr[i] = s * (1.0f - s);
    }
    __syncthreads();

    // ---- Main gather + dot loop: one wave per row ----
    const int lane   = tid & 31;
    const int waveIb = __builtin_amdgcn_readfirstlane(tid >> 5);  // wave-uniform SGPR
    const int wave   = blockIdx.x * WAVES_PER_BLOCK + waveIb;
    const int nWaves = gridDim.x * WAVES_PER_BLOCK;

    float lossAcc = 0.0f;
    for (int row = wave; row < B_ROWS; row += nWaves) {
        // wave-uniform -> scalar loads (s_load) of the four indices
        const int xi = x[row];
        const int yi = y[row];
        const int ri = r[row];
        const int li = l[row];

        const float4 a = *(const float4*)(Wx + (size_t)xi * H_DIM + (size_t)lane * 4);
        const float4 b = *(const float4*)(Wx + (size_t)yi * H_DIM + (size_t)lane * 4);
        const float4 c = *(const float4*)(sWr + ri * H_DIM + lane * 4);   // ds_load_b128

        float part = a.x * b.x * c.x;
        part = fmaf(a.y * b.y, c.y, part);
        part = fmaf(a.z * b.z, c.z, part);
        part = fmaf(a.w * b.w, c.w, part);

        // wave32 butterfly reduction
        #pragma unroll
        for (int off = 16; off > 0; off >>= 1)
            part += __shfl_xor(part, off, 32);

        if (lane == 0) {
            const float p = 1.0f / (1.0f + expf(-part));
            *(float2*)(out + 2 * (size_t)row) = make_float2(p, 1.0f - p);
            // -log_softmax([p, 1-p])[li]
            const float lse    = logf(expf(p) + expf(1.0f - p));
            const float picked = (li == 0) ? p : (1.0f - p);
            lossAcc += lse - picked;
        }
    }

    // ---- Deterministic per-block loss partial ----
    if (lane == 0) sWaveLoss[waveIb] = lossAcc;
    __syncthreads();
    if (tid == 0) {
        float s = 0.0f;
        #pragma unroll
        for (int w = 0; w < WAVES_PER_BLOCK; ++w) s += sWaveLoss[w];
        blockPartials[blockIdx.x] = s;
    }
}

// ---------------------------------------------------------------------------
// Kernel 2: deterministic fixed-order reduction of block partials -> mean loss
// ---------------------------------------------------------------------------
__global__ __launch_bounds__(BLOCK) void hin2vec_loss_finalize(
    const float* __restrict__ blockPartials, float* __restrict__ lossOut)
{
    __shared__ float sdata[BLOCK];
    const int tid = threadIdx.x;
    float s = 0.0f;
    for (int i = tid; i < NBLOCKS; i += BLOCK) s += blockPartials[i];
    sdata[tid] = s;
    __syncthreads();
    #pragma unroll
    for (int stride = BLOCK / 2; stride > 0; stride >>= 1) {
        if (tid < stride) sdata[tid] += sdata[tid + stride];
        __syncthreads();
    }
    if (tid == 0) lossOut[0] = sdata[0] * (1.0f / (float)B_ROWS);
}

// ---------------------------------------------------------------------------
extern "C" void kernel_launch(void* const* d_in, const int* in_sizes, int n_in,
                              void* d_out, int out_size, void* d_ws, size_t ws_size,
                              hipStream_t stream)
{
    const int*   x  = (const int*)d_in[0];
    const int*   y  = (const int*)d_in[1];
    const int*   r  = (const int*)d_in[2];
    const int*   l  = (const int*)d_in[3];
    const float* Wx = (const float*)d_in[4];
    const float* Wr = (const float*)d_in[5];

    float* out      = (float*)d_out;       // [B*2] logits, then [1] loss
    float* partials = (float*)d_ws;        // NBLOCKS floats of scratch

    hin2vec_fwd<<<NBLOCKS, BLOCK, 0, stream>>>(x, y, r, l, Wx, Wr, out, partials);
    hin2vec_loss_finalize<<<1, BLOCK, 0, stream>>>(partials, out + (size_t)2 * B_ROWS);
}